// LocalGraphLayer_2929167696095
// MI455X (gfx1250) — compile-verified
//
#include <hip/hip_runtime.h>

#define B_    64
#define K_    3
#define N_    8192
#define FIN   64
#define FOUT  64
#define DTOT  192                         // K_*FIN, contraction dim of epilogue GEMM
#define PER_BATCH_F4 ((K_ * N_ * FIN) / 4)  // 393216 float4 per batch
#define NB_PER_BATCH 16
#define RED_BLOCK    256

typedef __attribute__((ext_vector_type(2))) float v2f;
typedef __attribute__((ext_vector_type(4))) float v4f;
typedef __attribute__((ext_vector_type(8))) float v8f;

// ---------------------------------------------------------------------------
// Phase 0: zero the (B, 192) partial-sum buffer in workspace (ws is poisoned).
// ---------------------------------------------------------------------------
__global__ void zero_ws_kernel(float* __restrict__ s, int n) {
    int i = blockIdx.x * blockDim.x + threadIdx.x;
    if (i < n) s[i] = 0.0f;
}

// ---------------------------------------------------------------------------
// Phase 1: bandwidth-bound bucketed reduction.
// out S[b, j*64 + fin] = sum over p of x[b][p] where j=(p/64)%3, fin=p%64.
// Grid: (B_, NB_PER_BATCH), block RED_BLOCK. Streams 402.7 MB once (~17 us
// floor at 23.3 TB/s). float4 non-temporal loads; per-thread register
// accumulators for the 3 buckets (fin4 column is invariant per thread since
// the grid stride is a multiple of 16 float4s); LDS ds_add_f32 tree; then
// 192 global f32 atomics per block into d_ws.
// ---------------------------------------------------------------------------
__global__ __launch_bounds__(RED_BLOCK) void bucket_reduce_kernel(
        const float* __restrict__ x, float* __restrict__ S) {
    __shared__ float lsum[DTOT];
    const int tid = threadIdx.x;
    const int b   = blockIdx.x;

    if (tid < DTOT) lsum[tid] = 0.0f;
    __syncthreads();

    const v4f* xb = (const v4f*)x + (size_t)b * PER_BATCH_F4;
    const int q0     = blockIdx.y * RED_BLOCK + tid;
    const int stride = NB_PER_BATCH * RED_BLOCK;   // 4096, multiple of 16

    v4f a0 = (v4f)0.0f, a1 = (v4f)0.0f, a2 = (v4f)0.0f;

    for (int q = q0; q < PER_BATCH_F4; q += stride) {
        v4f v = __builtin_nontemporal_load(&xb[q]);   // streaming: NT hint
        const int kn = q >> 4;                        // 16 float4 per fin-row
        const int j  = kn % 3;
        const float w0 = (j == 0) ? 1.0f : 0.0f;
        const float w1 = (j == 1) ? 1.0f : 0.0f;
        const float w2 = (j == 2) ? 1.0f : 0.0f;
        a0 += w0 * v;   // v_fmac_f32 x4 per bucket, keeps acc in VGPRs
        a1 += w1 * v;
        a2 += w2 * v;
    }

    const int fin4 = (q0 & 15) * 4;   // column invariant across the q-loop
    atomicAdd(&lsum[0 * FIN + fin4 + 0], a0.x);
    atomicAdd(&lsum[0 * FIN + fin4 + 1], a0.y);
    atomicAdd(&lsum[0 * FIN + fin4 + 2], a0.z);
    atomicAdd(&lsum[0 * FIN + fin4 + 3], a0.w);
    atomicAdd(&lsum[1 * FIN + fin4 + 0], a1.x);
    atomicAdd(&lsum[1 * FIN + fin4 + 1], a1.y);
    atomicAdd(&lsum[1 * FIN + fin4 + 2], a1.z);
    atomicAdd(&lsum[1 * FIN + fin4 + 3], a1.w);
    atomicAdd(&lsum[2 * FIN + fin4 + 0], a2.x);
    atomicAdd(&lsum[2 * FIN + fin4 + 1], a2.y);
    atomicAdd(&lsum[2 * FIN + fin4 + 2], a2.z);
    atomicAdd(&lsum[2 * FIN + fin4 + 3], a2.w);
    __syncthreads();

    if (tid < DTOT) atomicAdd(&S[b * DTOT + tid], lsum[tid]);
}

// ---------------------------------------------------------------------------
// Phase 2: epilogue GEMM out = relu(S(64x192) @ Kr(192x64) + bias) using the
// fp32 WMMA path: V_WMMA_F32_16X16X4_F32. One workgroup, 16 waves (wave32),
// one 16x16 output tile per wave, 48 K-steps of 4. EXEC is all-ones (no
// divergence around the WMMA). Full fp32 precision, matching the reference.
// ISA layouts (7.12.2): A: lane l, vgpr r -> A[M=l%16, K=(l>>4)*2+r]
//                       B: lane l, vgpr r -> B[K=(l>>4)*2+r, N=l%16]
//                       D: lane l, vgpr r -> D[M=r+(l>>4)*8, N=l%16]
// ---------------------------------------------------------------------------
__global__ __launch_bounds__(512) void gemm_wmma_kernel(
        const float* __restrict__ S, const float* __restrict__ Kr,
        const float* __restrict__ bias, float* __restrict__ out) {
    const int lane  = threadIdx.x & 31;
    const int wv    = threadIdx.x >> 5;   // 0..15
    const int mt    = wv >> 2;            // 0..3 tile row
    const int nt    = wv & 3;             // 0..3 tile col
    const int row16 = lane & 15;
    const int half  = lane >> 4;          // 0 or 1

    v8f c = {};

    const float* Srow = S  + (mt * 16 + row16) * DTOT;   // A row for this lane
    const float* Bcol = Kr + nt * 16 + row16;            // B column for this lane

#pragma unroll 4
    for (int kb = 0; kb < DTOT; kb += 4) {
        const int ks = kb + half * 2;     // this lane's K slots: ks, ks+1
        v2f a, bb;
        a.x  = Srow[ks];                  // contiguous pair -> global_load_b64
        a.y  = Srow[ks + 1];
        bb.x = Bcol[(size_t)ks * FOUT];
        bb.y = Bcol[(size_t)(ks + 1) * FOUT];
        c = __builtin_amdgcn_wmma_f32_16x16x4_f32(
                /*neg_a=*/false, a, /*neg_b=*/false, bb,
                /*c_mod=*/(short)0, c, /*reuse_a=*/false, /*reuse_b=*/false);
    }

    const float bv = bias[nt * 16 + row16];
#pragma unroll
    for (int r = 0; r < 8; ++r) {
        const int M = mt * 16 + half * 8 + r;
        const int F = nt * 16 + row16;
        const float val = c[r] + bv;
        out[M * FOUT + F] = val > 0.0f ? val : 0.0f;   // fused bias + relu
    }
}

// ---------------------------------------------------------------------------
extern "C" void kernel_launch(void* const* d_in, const int* in_sizes, int n_in,
                              void* d_out, int out_size, void* d_ws, size_t ws_size,
                              hipStream_t stream) {
    const float* x    = (const float*)d_in[0];   // (B, K, N, F_IN) fp32
    const float* kern = (const float*)d_in[1];   // (K, F_IN, F_OUT) fp32 == (192, 64) row-major
    const float* bias = (const float*)d_in[2];   // (F_OUT,)
    float*       out  = (float*)d_out;           // (B, F_OUT)
    float*       S    = (float*)d_ws;            // (B, 192) partial sums, 48 KB

    zero_ws_kernel<<<(B_ * DTOT + 255) / 256, 256, 0, stream>>>(S, B_ * DTOT);

    dim3 grid(B_, NB_PER_BATCH);
    bucket_reduce_kernel<<<grid, RED_BLOCK, 0, stream>>>(x, S);

    gemm_wmma_kernel<<<1, 512, 0, stream>>>(S, kern, bias, out);
}